// PointsAggregation_27608049778805
// MI455X (gfx1250) — compile-verified
//
#include <hip/hip_runtime.h>
#include <hip/hip_bf16.h>

// ---------------- problem constants ----------------
#define BB 2
#define NN 16384
#define MM 4096
#define CIN 16
#define EPS 1e-5f

// ---------------- workspace layout (bytes) ----------------
// folded params region, float offsets:
#define FW_S0L0 0      // 16x20
#define FB_S0L0 320
#define FW_S0L1 336    // 16x16
#define FB_S0L1 592
#define FW_S0L2 608    // 32x16
#define FB_S0L2 1120
#define FW_S1L0 1152   // 32x20
#define FB_S1L0 1792
#define FW_S1L1 1824   // 32x32
#define FB_S1L1 2848
#define FW_S1L2 2880   // 64x32
#define FB_S1L2 4928
#define FW_AGG  4992   // 64x96
#define FB_AGG  11136
#define PARAM_FLOATS 11200

#define IDX0_OFF 45056                       // int32, B*M*16
#define IDX1_OFF (IDX0_OFF + BB*MM*16*4)     // int32, B*M*32
#define P0_OFF   (IDX1_OFF + BB*MM*32*4)     // f32, B*32*M
#define P1_OFF   (P0_OFF   + BB*32*MM*4)     // f32, B*64*M

typedef float v2f __attribute__((ext_vector_type(2)));
typedef float v8f __attribute__((ext_vector_type(8)));

static __device__ __forceinline__ v8f wmma4(v2f a, v2f b, v8f c) {
  // D(16x16,f32) = A(16x4,f32) * B(4x16,f32) + C   -> v_wmma_f32_16x16x4_f32
  return __builtin_amdgcn_wmma_f32_16x16x4_f32(false, a, false, b, (short)0, c,
                                               false, false);
}

// ---------------- param folding: W' = (g/sqrt(v+eps))*W (zero-padded K),
//                  b' = (b-mean)*(g/sqrt(v+eps)) + beta ----------------
__global__ void fold_kernel(const float* __restrict__ w, const float* __restrict__ b,
                            const float* __restrict__ g, const float* __restrict__ beta,
                            const float* __restrict__ mean, const float* __restrict__ var,
                            float* __restrict__ fw, float* __restrict__ fb,
                            int cout, int cin, int cinP) {
  int t = blockIdx.x * blockDim.x + threadIdx.x;
  if (t < cout) {
    float a = g[t] * rsqrtf(var[t] + EPS);
    fb[t] = (b[t] - mean[t]) * a + beta[t];
  }
  for (int i = t; i < cout * cinP; i += blockDim.x * gridDim.x) {
    int o = i / cinP, c = i - o * cinP;
    float a = g[o] * rsqrtf(var[o] + EPS);
    fw[i] = (c < cin) ? w[o * cin + c] * a : 0.0f;
  }
}

// ---------------- ball query: one wave32 per centroid, ballot-ordered compaction ----
// Preserves reference semantics: first-k in-ball points in ascending index order,
// padded with the first hit (0 if the ball is empty). Both radii in one scan.
__global__ void __launch_bounds__(256) ballquery_kernel(const float* __restrict__ pts,
                                                        const float* __restrict__ cents,
                                                        int* __restrict__ idx0,
                                                        int* __restrict__ idx1) {
  const int wave = threadIdx.x >> 5;
  const int lane = threadIdx.x & 31;
  const int g = blockIdx.x * 8 + wave;   // centroid id 0 .. B*M-1
  const int b = g >> 12;                 // / M
  const int m = g & (MM - 1);
  const float r0sq = 0.8f * 0.8f;
  const float r1sq = 1.6f * 1.6f;
  const float* P = pts + (size_t)b * NN * 3;
  const float cx = cents[((size_t)b * MM + m) * 3 + 0];
  const float cy = cents[((size_t)b * MM + m) * 3 + 1];
  const float cz = cents[((size_t)b * MM + m) * 3 + 2];
  int* o0 = idx0 + (size_t)g * 16;
  int* o1 = idx1 + (size_t)g * 32;
  int c0 = 0, c1 = 0, first0 = 0, first1 = 0;   // all wave-uniform
  const unsigned ltmask = (1u << lane) - 1u;
  for (int i0 = 0; i0 < NN; i0 += 32) {
    if (c0 >= 16 && c1 >= 32) break;            // uniform branch
    const int i = i0 + lane;
    const float dx = P[3 * i + 0] - cx;
    const float dy = P[3 * i + 1] - cy;
    const float dz = P[3 * i + 2] - cz;
    const float d2 = dx * dx + dy * dy + dz * dz;
    const bool h0 = d2 < r0sq;
    const bool h1 = d2 < r1sq;
    const unsigned bal0 = (unsigned)__ballot(h0);
    const unsigned bal1 = (unsigned)__ballot(h1);
    if (c0 < 16) {
      if (c0 == 0 && bal0) first0 = i0 + __builtin_ctz(bal0);
      const int pos = c0 + __popc(bal0 & ltmask);
      if (h0 && pos < 16) o0[pos] = i;
      c0 = min(16, c0 + __popc(bal0));
    }
    if (c1 < 32) {
      if (c1 == 0 && bal1) first1 = i0 + __builtin_ctz(bal1);
      const int pos = c1 + __popc(bal1 & ltmask);
      if (h1 && pos < 32) o1[pos] = i;
      c1 = min(32, c1 + __popc(bal1));
    }
  }
  // pad unfilled slots with first hit (wave-uniform values)
  if (lane >= c0 && lane < 16) o0[lane] = first0;
  if (lane >= c1) o1[lane] = first1;
}

// ---------------- WMMA GEMM helpers ----------------
// A layout (16x4 f32):  M = lane&15 ; K = r + 2*(lane>=16)  (r = vgpr 0/1)
// B layout (4x16 f32):  N = lane&15 ; K = r + 2*(lane>=16)
// D layout (16x16 f32): N = lane&15 ; M = r + 8*(lane>=16)  (r = vgpr 0..7)
template <int CINP>
static __device__ __forceinline__ v8f gemm_tile(const float* __restrict__ W, int rowBase,
                                                const float* X, int lane) {
  v8f acc = {0.f, 0.f, 0.f, 0.f, 0.f, 0.f, 0.f, 0.f};
  const int col = lane & 15;
  const int koff = (lane >> 4) * 2;
#pragma unroll
  for (int k0 = 0; k0 < CINP; k0 += 4) {
    v2f a, bv;
    a[0] = W[(rowBase + col) * CINP + k0 + koff + 0];
    a[1] = W[(rowBase + col) * CINP + k0 + koff + 1];
    bv[0] = X[(k0 + koff + 0) * 16 + col];
    bv[1] = X[(k0 + koff + 1) * 16 + col];
    acc = wmma4(a, bv, acc);
  }
  return acc;
}

static __device__ __forceinline__ void store_relu(v8f acc, const float* __restrict__ bias,
                                                  int base, float* Y, int lane) {
  const int col = lane & 15;
  const int rowOff = (lane >> 4) * 8;
#pragma unroll
  for (int r = 0; r < 8; ++r) {
    int row = base + r + rowOff;
    float y = acc[r] + bias[row];
    y = y > 0.f ? y : 0.f;
    Y[row * 16 + col] = y;
  }
}

// ---------------- fused gather + 3-layer MLP + max-pool, one centroid per wave ----------------
template <int KNEI, int C1, int C2, int C3>
__global__ void __launch_bounds__(256) msg_kernel(
    const float* __restrict__ pts, const float* __restrict__ feats,
    const float* __restrict__ cents, const int* __restrict__ idx,
    const float* __restrict__ fold, int fw0, int fb0, int fw1, int fb1, int fw2, int fb2,
    float* __restrict__ pooled) {
  __shared__ float lds[8][2][32 * 16];  // per-wave ping-pong buffers (32 rows x 16 cols)
  const int wave = threadIdx.x >> 5;
  const int lane = threadIdx.x & 31;
  const int g = blockIdx.x * 8 + wave;   // centroid id 0 .. B*M-1
  const int b = g >> 12;
  const int m = g & (MM - 1);
  float* bufA = lds[wave][0];
  float* bufB = lds[wave][1];
  const float* W1 = fold + fw0; const float* B1 = fold + fb0;
  const float* W2 = fold + fw1; const float* B2 = fold + fb1;
  const float* W3 = fold + fw2; const float* B3 = fold + fb2;
  const int* myidx = idx + (size_t)g * KNEI;
  const float cx = cents[((size_t)b * MM + m) * 3 + 0];
  const float cy = cents[((size_t)b * MM + m) * 3 + 1];
  const float cz = cents[((size_t)b * MM + m) * 3 + 2];

  float mx[C3 / 16][8];
#pragma unroll
  for (int rt = 0; rt < C3 / 16; ++rt)
#pragma unroll
    for (int r = 0; r < 8; ++r) mx[rt][r] = -3.402823e38f;

  const int col = lane & 15;
  const int half = lane >> 4;

  for (int j = 0; j < KNEI / 16; ++j) {
    // ---- build X (rows: 0-2 rel xyz, 3-18 features, 19 pad) into bufA ----
    int pi = myidx[j * 16 + col];
    float px = pts[((size_t)b * NN + pi) * 3 + 0] - cx;
    float py = pts[((size_t)b * NN + pi) * 3 + 1] - cy;
    float pz = pts[((size_t)b * NN + pi) * 3 + 2] - cz;
    if (lane < 16) {
      bufA[0 * 16 + col] = px;
      bufA[1 * 16 + col] = py;
      bufA[2 * 16 + col] = pz;
      bufA[19 * 16 + col] = 0.f;
    }
#pragma unroll
    for (int c = 0; c < 8; ++c) {
      int ch = c + half * 8;
      bufA[(3 + ch) * 16 + col] = feats[((size_t)b * CIN + ch) * NN + pi];
    }
    __syncthreads();
    // ---- layer 1: 20 -> C1 ----
#pragma unroll
    for (int rt = 0; rt < C1 / 16; ++rt) {
      v8f acc = gemm_tile<20>(W1, rt * 16, bufA, lane);
      store_relu(acc, B1, rt * 16, bufB, lane);
    }
    __syncthreads();
    // ---- layer 2: C1 -> C2 ----
#pragma unroll
    for (int rt = 0; rt < C2 / 16; ++rt) {
      v8f acc = gemm_tile<C1>(W2, rt * 16, bufB, lane);
      store_relu(acc, B2, rt * 16, bufA, lane);
    }
    __syncthreads();
    // ---- layer 3: C2 -> C3, fused column-max ----
#pragma unroll
    for (int rt = 0; rt < C3 / 16; ++rt) {
      v8f acc = gemm_tile<C2>(W3, rt * 16, bufA, lane);
#pragma unroll
      for (int r = 0; r < 8; ++r) {
        int ch = rt * 16 + r + half * 8;
        float y = acc[r] + B3[ch];
        y = y > 0.f ? y : 0.f;
#pragma unroll
        for (int msk = 1; msk < 16; msk <<= 1)
          y = fmaxf(y, __shfl_xor(y, msk, 16));
        mx[rt][r] = fmaxf(mx[rt][r], y);
      }
    }
    __syncthreads();
  }
  // ---- write pooled (B, C3, M); lanes 0 and 16 hold group maxima ----
  if ((lane & 15) == 0) {
#pragma unroll
    for (int rt = 0; rt < C3 / 16; ++rt)
#pragma unroll
      for (int r = 0; r < 8; ++r) {
        int ch = rt * 16 + r + half * 8;
        pooled[((size_t)b * C3 + ch) * MM + m] = mx[rt][r];
      }
  }
}

// ---------------- aggregation layer: 64x96 GEMM over concat(pooled0,pooled1) ----------------
__global__ void __launch_bounds__(256) agg_kernel(const float* __restrict__ p0,
                                                  const float* __restrict__ p1,
                                                  const float* __restrict__ fold,
                                                  float* __restrict__ out) {
  const int wave = threadIdx.x >> 5;
  const int lane = threadIdx.x & 31;
  const int tile = blockIdx.x * 8 + wave;   // 0 .. B*M/16-1
  const int b = tile >> 8;                  // M/16 = 256 tiles per batch
  const int m0 = (tile & 255) * 16;
  const int col = lane & 15;
  const int koff = (lane >> 4) * 2;
  const float* W = fold + FW_AGG;
  const float* Bb = fold + FB_AGG;
  v8f acc[4];
#pragma unroll
  for (int rt = 0; rt < 4; ++rt) acc[rt] = (v8f){0.f, 0.f, 0.f, 0.f, 0.f, 0.f, 0.f, 0.f};
#pragma unroll
  for (int k0 = 0; k0 < 96; k0 += 4) {
    v2f bv;
#pragma unroll
    for (int r = 0; r < 2; ++r) {
      int ch = k0 + koff + r;
      bv[r] = (ch < 32) ? p0[((size_t)b * 32 + ch) * MM + m0 + col]
                        : p1[((size_t)b * 64 + (ch - 32)) * MM + m0 + col];
    }
#pragma unroll
    for (int rt = 0; rt < 4; ++rt) {
      v2f a;
      a[0] = W[(rt * 16 + col) * 96 + k0 + koff + 0];
      a[1] = W[(rt * 16 + col) * 96 + k0 + koff + 1];
      acc[rt] = wmma4(a, bv, acc[rt]);
    }
  }
  const int rowOff = (lane >> 4) * 8;
#pragma unroll
  for (int rt = 0; rt < 4; ++rt)
#pragma unroll
    for (int r = 0; r < 8; ++r) {
      int ch = rt * 16 + r + rowOff;
      float y = acc[rt][r] + Bb[ch];
      y = y > 0.f ? y : 0.f;
      out[((size_t)b * 64 + ch) * MM + m0 + col] = y;
    }
}

// ---------------- host ----------------
extern "C" void kernel_launch(void* const* d_in, const int* in_sizes, int n_in,
                              void* d_out, int out_size, void* d_ws, size_t ws_size,
                              hipStream_t stream) {
  (void)in_sizes; (void)n_in; (void)out_size; (void)ws_size;
  const float* points   = (const float*)d_in[0];
  const float* features = (const float*)d_in[1];
  const float* cents    = (const float*)d_in[2];
  char* ws = (char*)d_ws;
  float* fold = (float*)ws;
  int* idx0 = (int*)(ws + IDX0_OFF);
  int* idx1 = (int*)(ws + IDX1_OFF);
  float* pooled0 = (float*)(ws + P0_OFF);
  float* pooled1 = (float*)(ws + P1_OFF);
  float* out = (float*)d_out;

  // fold BN into weights/bias: {in_base, fw_off, fb_off, cout, cin, cinP}
  const int tab[7][6] = {
      {3,  FW_S0L0, FB_S0L0, 16, 19, 20},
      {9,  FW_S0L1, FB_S0L1, 16, 16, 16},
      {15, FW_S0L2, FB_S0L2, 32, 16, 16},
      {21, FW_S1L0, FB_S1L0, 32, 19, 20},
      {27, FW_S1L1, FB_S1L1, 32, 32, 32},
      {33, FW_S1L2, FB_S1L2, 64, 32, 32},
      {39, FW_AGG,  FB_AGG,  64, 96, 96},
  };
  for (int l = 0; l < 7; ++l) {
    int base = tab[l][0];
    int total = tab[l][3] * tab[l][5];
    int blocks = (total + 255) / 256;
    fold_kernel<<<blocks, 256, 0, stream>>>(
        (const float*)d_in[base + 0], (const float*)d_in[base + 1],
        (const float*)d_in[base + 2], (const float*)d_in[base + 3],
        (const float*)d_in[base + 4], (const float*)d_in[base + 5],
        fold + tab[l][1], fold + tab[l][2], tab[l][3], tab[l][4], tab[l][5]);
  }

  // ball query: one wave per centroid, ballot-ordered compaction
  ballquery_kernel<<<(BB * MM) / 8, 256, 0, stream>>>(points, cents, idx0, idx1);

  // scale 0: k=16, 19->16->16->32
  msg_kernel<16, 16, 16, 32><<<(BB * MM) / 8, 256, 0, stream>>>(
      points, features, cents, idx0, fold,
      FW_S0L0, FB_S0L0, FW_S0L1, FB_S0L1, FW_S0L2, FB_S0L2, pooled0);

  // scale 1: k=32, 19->32->32->64
  msg_kernel<32, 32, 32, 64><<<(BB * MM) / 8, 256, 0, stream>>>(
      points, features, cents, idx1, fold,
      FW_S1L0, FB_S1L0, FW_S1L1, FB_S1L1, FW_S1L2, FB_S1L2, pooled1);

  // aggregation: 96 -> 64
  agg_kernel<<<(BB * MM / 16) / 8, 256, 0, stream>>>(pooled0, pooled1, fold, out);
}